// JacobianMLP_17360257810985
// MI455X (gfx1250) — compile-verified
//
#include <hip/hip_runtime.h>
#include <hip/hip_bf16.h>
#include <cstddef>

// ---------------------------------------------------------------------------
// Sizes (fixed by the reference): insize=2048, hsize=4096, outsize=2048
// ---------------------------------------------------------------------------
#define INSIZE  2048
#define HSIZE   4096
#define OUTSIZE 2048

typedef float v2f __attribute__((ext_vector_type(2)));
typedef float v8f __attribute__((ext_vector_type(8)));

// ---------------------------------------------------------------------------
// GEMV: h[row] = relu(W[row,:] . v),  m[row] = (z>0);  if m==nullptr, h = z.
// One block per row, 256 threads, float4 loads, LDS tree reduction.
// ---------------------------------------------------------------------------
__global__ __launch_bounds__(256) void gemv_relu_kernel(
    const float* __restrict__ W, const float* __restrict__ v,
    float* __restrict__ h, float* __restrict__ m, int cols)
{
  __shared__ float red[256];
  const int row = blockIdx.x;
  const float* wr = W + (size_t)row * cols;
  float s = 0.f;
  for (int c = threadIdx.x * 4; c < cols; c += 256 * 4) {
    float4 wv = *(const float4*)&wr[c];
    float4 xv = *(const float4*)&v[c];
    s += wv.x * xv.x + wv.y * xv.y + wv.z * xv.z + wv.w * xv.w;
  }
  red[threadIdx.x] = s;
  __syncthreads();
  for (int off = 128; off > 0; off >>= 1) {
    if (threadIdx.x < off) red[threadIdx.x] += red[threadIdx.x + off];
    __syncthreads();
  }
  if (threadIdx.x == 0) {
    float z = red[0];
    if (m) {
      m[row] = (z > 0.f) ? 1.f : 0.f;
      h[row] = (z > 0.f) ? z : 0.f;
    } else {
      h[row] = z;
    }
  }
}

// ---------------------------------------------------------------------------
// Transpose: Out[C,R] = In[R,C]^T.  32x32 LDS tile, 32x8 threads.
// ---------------------------------------------------------------------------
__global__ __launch_bounds__(256) void transpose_kernel(
    const float* __restrict__ In, float* __restrict__ Out, int R, int C)
{
  __shared__ float tile[32][33];
  const int cBase = blockIdx.x * 32;
  const int rBase = blockIdx.y * 32;
  const int tx = threadIdx.x, ty = threadIdx.y;
  #pragma unroll
  for (int i = 0; i < 32; i += 8)
    tile[ty + i][tx] = In[(size_t)(rBase + ty + i) * C + (cBase + tx)];
  __syncthreads();
  #pragma unroll
  for (int i = 0; i < 32; i += 8)
    Out[(size_t)(cBase + ty + i) * R + (rBase + tx)] = tile[tx][ty + i];
}

// ---------------------------------------------------------------------------
// Diagonal fill: Out[n,n] = diag(m) (or eye(n) when m==nullptr). float4 stores.
// ---------------------------------------------------------------------------
__global__ __launch_bounds__(256) void diag_kernel(
    float* __restrict__ Out, const float* __restrict__ m, int n)
{
  size_t g = (size_t)blockIdx.x * 256 + threadIdx.x;
  size_t base = g * 4;
  if (base >= (size_t)n * n) return;
  int row = (int)(base / (size_t)n);
  int col = (int)(base - (size_t)row * n);
  float4 val = {0.f, 0.f, 0.f, 0.f};
  int d = row - col;                 // diagonal lands in this float4 iff 0<=d<4
  if (d >= 0 && d < 4) ((float*)&val)[d] = m ? m[row] : 1.0f;
  *(float4*)&Out[base] = val;
}

// ---------------------------------------------------------------------------
// FP32 WMMA GEMM:  C[M,N] = A[M,K] @ (rowScale ⊙rows B[K,N])
//   storeT==0 : C[row*N + col]
//   storeT==1 : C[col*M + row]     (transposed store, used for DJM)
// Block tile 128x128, BK=16, 8 wave32s (4x2), wave tile 32x64 => 2x4 v8f acc.
// A staged K-major in LDS (AlsT[k][m]), B staged [k][n]; both padded to 136
// floats/row so fragment reads are bank-conflict-free (2 rows apart = +16 banks).
// V_WMMA_F32_16X16X4_F32 per ISA 7.12.2 lane layouts.
// ---------------------------------------------------------------------------
#define BM 128
#define BN 128
#define BK 16
#define LPAD 136

__global__ __launch_bounds__(256) void gemm_f32_wmma_kernel(
    const float* __restrict__ A, int lda,
    const float* __restrict__ B, int ldb,
    const float* __restrict__ rowScale,
    float* __restrict__ C, int M, int N, int K, int storeT)
{
  __shared__ float AlsT[BK][LPAD];   // [k][m] within tile
  __shared__ float Bls [BK][LPAD];   // [k][n] within tile

  const int t    = threadIdx.x;
  const int lane = t & 31;
  const int wave = t >> 5;
  const int wm   = wave >> 1;        // 0..3  (M direction)
  const int wn   = wave & 1;         // 0..1  (N direction)
  const int blockRow = blockIdx.y * BM;
  const int blockCol = blockIdx.x * BN;

  const v8f vzero = {0.f,0.f,0.f,0.f,0.f,0.f,0.f,0.f};
  v8f acc[2][4];
  #pragma unroll
  for (int i = 0; i < 2; ++i)
    #pragma unroll
    for (int j = 0; j < 4; ++j) acc[i][j] = vzero;

  // Per-thread staging indices: A tile is 128x16 (2 float4/thread),
  // B tile is 16x128 (2 float4/thread).
  const int ar0 = t >> 2;            // 0..63 (+64 for second load)
  const int ac0 = (t & 3) * 4;       // 0,4,8,12
  const int br0 = t >> 5;            // 0..7  (+8 for second load)
  const int bc0 = (t & 31) * 4;      // 0..124

  float4 aReg0, aReg1, bReg0, bReg1;
  float  s0 = 1.f, s1 = 1.f;

  auto loadStage = [&](int k0) {
    aReg0 = *(const float4*)&A[(size_t)(blockRow + ar0)      * lda + k0 + ac0];
    aReg1 = *(const float4*)&A[(size_t)(blockRow + ar0 + 64) * lda + k0 + ac0];
    bReg0 = *(const float4*)&B[(size_t)(k0 + br0)     * ldb + blockCol + bc0];
    bReg1 = *(const float4*)&B[(size_t)(k0 + br0 + 8) * ldb + blockCol + bc0];
    s0 = rowScale ? rowScale[k0 + br0]     : 1.f;
    s1 = rowScale ? rowScale[k0 + br0 + 8] : 1.f;
  };
  auto storeStage = [&]() {
    AlsT[ac0 + 0][ar0]      = aReg0.x;
    AlsT[ac0 + 1][ar0]      = aReg0.y;
    AlsT[ac0 + 2][ar0]      = aReg0.z;
    AlsT[ac0 + 3][ar0]      = aReg0.w;
    AlsT[ac0 + 0][ar0 + 64] = aReg1.x;
    AlsT[ac0 + 1][ar0 + 64] = aReg1.y;
    AlsT[ac0 + 2][ar0 + 64] = aReg1.z;
    AlsT[ac0 + 3][ar0 + 64] = aReg1.w;
    float4 b0 = make_float4(bReg0.x * s0, bReg0.y * s0, bReg0.z * s0, bReg0.w * s0);
    float4 b1 = make_float4(bReg1.x * s1, bReg1.y * s1, bReg1.z * s1, bReg1.w * s1);
    *(float4*)&Bls[br0][bc0]     = b0;
    *(float4*)&Bls[br0 + 8][bc0] = b1;
  };

  const int r  = lane & 15;          // row/col within 16
  const int hf = lane >> 4;          // lane half selects K pair {0,1} vs {2,3}

  loadStage(0);
  for (int k0 = 0; k0 < K; k0 += BK) {
    storeStage();
    __syncthreads();
    if (k0 + BK < K) loadStage(k0 + BK);

    #pragma unroll
    for (int kk = 0; kk < BK; kk += 4) {
      v2f a0, a1, b[4];
      // A 16x4 layout: lanes0-15 -> {K0,K1}, lanes16-31 -> {K2,K3}
      a0.x = AlsT[kk + 2 * hf    ][wm * 32 + r];
      a0.y = AlsT[kk + 2 * hf + 1][wm * 32 + r];
      a1.x = AlsT[kk + 2 * hf    ][wm * 32 + 16 + r];
      a1.y = AlsT[kk + 2 * hf + 1][wm * 32 + 16 + r];
      // B 4x16 layout: VGPR0 = rows {2*hf}, VGPR1 = rows {2*hf+1}
      #pragma unroll
      for (int j = 0; j < 4; ++j) {
        b[j].x = Bls[kk + 2 * hf    ][wn * 64 + j * 16 + r];
        b[j].y = Bls[kk + 2 * hf + 1][wn * 64 + j * 16 + r];
      }
      acc[0][0] = __builtin_amdgcn_wmma_f32_16x16x4_f32(false, a0, false, b[0], (short)0, acc[0][0], false, false);
      acc[0][1] = __builtin_amdgcn_wmma_f32_16x16x4_f32(false, a0, false, b[1], (short)0, acc[0][1], false, false);
      acc[0][2] = __builtin_amdgcn_wmma_f32_16x16x4_f32(false, a0, false, b[2], (short)0, acc[0][2], false, false);
      acc[0][3] = __builtin_amdgcn_wmma_f32_16x16x4_f32(false, a0, false, b[3], (short)0, acc[0][3], false, false);
      acc[1][0] = __builtin_amdgcn_wmma_f32_16x16x4_f32(false, a1, false, b[0], (short)0, acc[1][0], false, false);
      acc[1][1] = __builtin_amdgcn_wmma_f32_16x16x4_f32(false, a1, false, b[1], (short)0, acc[1][1], false, false);
      acc[1][2] = __builtin_amdgcn_wmma_f32_16x16x4_f32(false, a1, false, b[2], (short)0, acc[1][2], false, false);
      acc[1][3] = __builtin_amdgcn_wmma_f32_16x16x4_f32(false, a1, false, b[3], (short)0, acc[1][3], false, false);
    }
    __syncthreads();
  }

  // C/D layout: VGPR v holds (M=v, lanes0-15) and (M=v+8, lanes16-31), N=lane%16
  #pragma unroll
  for (int i = 0; i < 2; ++i) {
    const int rowBase = blockRow + wm * 32 + i * 16 + hf * 8;
    #pragma unroll
    for (int j = 0; j < 4; ++j) {
      const int col = blockCol + wn * 64 + j * 16 + r;
      #pragma unroll
      for (int v = 0; v < 8; ++v) {
        const int row = rowBase + v;
        if (storeT) C[(size_t)col * M + row] = acc[i][j][v];
        else        C[(size_t)row * N + col] = acc[i][j][v];
      }
    }
  }
}

// ---------------------------------------------------------------------------
// Launch: output tuple layout (flat, return order):
//   out[2048] | DJM[2048*2048] | W1^T[2048*4096] | diag(m1)[4096^2]
//   | W2^T[4096^2] | diag(m2)[4096^2] | W3^T[4096*2048] | eye(2048)
// ---------------------------------------------------------------------------
extern "C" void kernel_launch(void* const* d_in, const int* in_sizes, int n_in,
                              void* d_out, int out_size, void* d_ws, size_t ws_size,
                              hipStream_t stream) {
  const float* x  = (const float*)d_in[0];   // [1, 2048]
  const float* W1 = (const float*)d_in[1];   // [4096, 2048]
  const float* W2 = (const float*)d_in[2];   // [4096, 4096]
  const float* W3 = (const float*)d_in[3];   // [2048, 4096]
  float* out = (float*)d_out;

  // workspace
  float* ws = (float*)d_ws;
  float* m1 = ws;                 // 4096
  float* h1 = ws + 4096;          // 4096
  float* m2 = ws + 8192;          // 4096
  float* h2 = ws + 12288;         // 4096
  float* V  = ws + 16384;         // [4096, 2048] intermediate (32 MB)

  const size_t OFF_OUT = 0;
  const size_t OFF_DJM = 2048;
  const size_t OFF_J1  = OFF_DJM + (size_t)OUTSIZE * OUTSIZE;          // 4196352
  const size_t OFF_J2  = OFF_J1  + (size_t)INSIZE  * HSIZE;            // 12584960
  const size_t OFF_J3  = OFF_J2  + (size_t)HSIZE   * HSIZE;            // 29362176
  const size_t OFF_J4  = OFF_J3  + (size_t)HSIZE   * HSIZE;            // 46139392
  const size_t OFF_J5  = OFF_J4  + (size_t)HSIZE   * HSIZE;            // 62916608
  const size_t OFF_J6  = OFF_J5  + (size_t)HSIZE   * OUTSIZE;          // 71305216

  // 1) GEMV chain (also produces ReLU masks)
  gemv_relu_kernel<<<HSIZE,   256, 0, stream>>>(W1, x,  h1, m1, INSIZE);
  gemv_relu_kernel<<<HSIZE,   256, 0, stream>>>(W2, h1, h2, m2, HSIZE);
  gemv_relu_kernel<<<OUTSIZE, 256, 0, stream>>>(W3, h2, out + OFF_OUT, nullptr, HSIZE);

  // 2) Jacobian tuple outputs (transposes + diagonal fills)
  {
    dim3 b(32, 8);
    transpose_kernel<<<dim3(INSIZE / 32, HSIZE / 32), b, 0, stream>>>(W1, out + OFF_J1, HSIZE, INSIZE);
    transpose_kernel<<<dim3(HSIZE  / 32, HSIZE / 32), b, 0, stream>>>(W2, out + OFF_J3, HSIZE, HSIZE);
    transpose_kernel<<<dim3(HSIZE  / 32, OUTSIZE / 32), b, 0, stream>>>(W3, out + OFF_J5, OUTSIZE, HSIZE);
  }
  {
    int gH = (int)(((size_t)HSIZE * HSIZE / 4 + 255) / 256);
    int gO = (int)(((size_t)OUTSIZE * OUTSIZE / 4 + 255) / 256);
    diag_kernel<<<gH, 256, 0, stream>>>(out + OFF_J2, m1, HSIZE);
    diag_kernel<<<gH, 256, 0, stream>>>(out + OFF_J4, m2, HSIZE);
    diag_kernel<<<gO, 256, 0, stream>>>(out + OFF_J6, nullptr, OUTSIZE);
  }

  // 3) DJM via two fp32 WMMA GEMMs:
  //    V     = W2 @ (m1 ⊙rows W1)            [4096, 2048]
  //    DJM^T = W3 @ (m2 ⊙rows V)             [2048, 2048], stored transposed
  gemm_f32_wmma_kernel<<<dim3(INSIZE / BN, HSIZE / BM), 256, 0, stream>>>(
      W2, HSIZE, W1, INSIZE, m1, V, HSIZE, INSIZE, HSIZE, 0);
  gemm_f32_wmma_kernel<<<dim3(OUTSIZE / BN, OUTSIZE / BM), 256, 0, stream>>>(
      W3, HSIZE, V, INSIZE, m2, out + OFF_DJM, OUTSIZE, OUTSIZE, HSIZE, 1);
}